// AngularBasis_89240830476882
// MI455X (gfx1250) — compile-verified
//
#include <hip/hip_runtime.h>
#include <math.h>

typedef float v2f __attribute__((ext_vector_type(2)));
typedef float v8f __attribute__((ext_vector_type(8)));

#define LMAX  7
#define NOUT  64   // (LMAX+1)^2 outputs per point
#define NFEAT 64   // monomial features per point

// Feature index helpers.
// c-features: m=0..7, k=0..7-m  -> flat 0..35
// s-features: m=1..7, k=0..7-m  -> flat 36..63
__device__ __host__ __forceinline__ constexpr int fcIdx(int m, int k) {
    return m * 8 - (m * (m - 1)) / 2 + k;
}
__device__ __host__ __forceinline__ constexpr int fsIdx(int m, int k) {
    return 36 + (m - 1) * 8 - (m * (m - 1)) / 2 + k;
}
// Inverse map: flat feature f -> (m, k). Folds to constants after unrolling.
__device__ __forceinline__ constexpr int featMK(int f, int wantK) {
    int ff = (f < 36) ? f : (f - 36);
    int m  = (f < 36) ? 0 : 1;
    int off = 0;
    while (off + (8 - m) <= ff) { off += (8 - m); ++m; }
    return wantK ? (ff - off) : m;
}
#define FM(f) featMK((f), 0)
#define FK(f) featMK((f), 1)

// ---------------------------------------------------------------------------
// Init kernel: W[64 feat][64 out] + per-output store-address meta in d_ws.
// One thread per output column n = l*l + l + m; associated-Legendre recurrence
// on z-polynomial coefficients in double precision.
// ---------------------------------------------------------------------------
__global__ void sh_init_W(float* __restrict__ W) {
    int n = threadIdx.x;
    if (n >= NOUT) return;

    for (int k = 0; k < NFEAT; ++k) W[k * NOUT + n] = 0.0f;

    int l = 0;
    while ((l + 1) * (l + 1) <= n) ++l;
    int m  = n - l * l - l;
    int am = m < 0 ? -m : m;

    // store-addressing meta: {l*l, 2l+1, n-l*l, 0}
    int4* meta = (int4*)(W + NFEAT * NOUT);
    meta[n] = make_int4(l * l, 2 * l + 1, n - l * l, 0);

    // (2*am-1)!!
    double dfact = 1.0;
    for (int i = 0; i < am; ++i) dfact *= (double)(2 * i + 1);

    // Q[l][am](z) polynomial coefficients:
    //   Q[am][am] = dfact
    //   Q[ll][am] = ((2ll-1) z Q[ll-1][am] - (ll+am-1) Q[ll-2][am]) / (ll-am)
    double cur[8], prev[8], nxt[8];
    for (int k = 0; k < 8; ++k) { cur[k] = 0.0; prev[k] = 0.0; }
    cur[0] = dfact;
    for (int ll = am + 1; ll <= l; ++ll) {
        for (int k = 0; k < 8; ++k) {
            double v = 0.0;
            if (k >= 1) v += (double)(2 * ll - 1) * cur[k - 1];
            v -= (double)(ll + am - 1) * prev[k];
            nxt[k] = v / (double)(ll - am);
        }
        for (int k = 0; k < 8; ++k) { prev[k] = cur[k]; cur[k] = nxt[k]; }
    }

    double fnum = 1.0, fden = 1.0;
    for (int i = 2; i <= l - am; ++i) fnum *= (double)i;
    for (int i = 2; i <= l + am; ++i) fden *= (double)i;
    double f = sqrt((double)(2 * l + 1) / (4.0 * M_PI) * fnum / fden);
    if (am != 0) f *= sqrt(2.0);

    for (int k = 0; k <= LMAX - am; ++k) {
        int row = (m >= 0) ? fcIdx(am, k) : fsIdx(am, k);
        W[row * NOUT + n] = (float)(f * cur[k]);
    }
}

// ---------------------------------------------------------------------------
// Main kernel: 256 threads = 8 waves, each wave owns 16 points.
// FEAT (per-lane VALU, never materialized as an array) x W (LDS) via
// V_WMMA_F32_16X16X4_F32, K=64 in 16 steps, 4 output tiles.
// ---------------------------------------------------------------------------
__global__ __launch_bounds__(256) void sh_wmma_kernel(
    const float* __restrict__ xyz, const float* __restrict__ W,
    float* __restrict__ out, int Np) {

    __shared__ float sW[NFEAT * NOUT];   // 16 KB
    __shared__ int4  sMeta[NOUT];        // 1 KB
    {
        const float4* Wg = (const float4*)W;
        float4*       Ws = (float4*)sW;
        #pragma unroll
        for (int i = threadIdx.x; i < (NFEAT * NOUT) / 4; i += 256) Ws[i] = Wg[i];
        if (threadIdx.x < NOUT)
            sMeta[threadIdx.x] = ((const int4*)(W + NFEAT * NOUT))[threadIdx.x];
    }
    __syncthreads();

    const int  tid  = threadIdx.x;
    const int  lane = tid & 31;        // wave32
    const int  wave = tid >> 5;
    const int  col  = lane & 15;       // point-in-tile / output-in-tile
    const int  h    = lane >> 4;       // lane half (K split per ISA layout)
    const bool hb   = (h != 0);
    const int  wb   = blockIdx.x * 128 + wave * 16;

    // --- per-lane monomial ingredients (both halves: same point) ----------
    int p  = wb + col;
    int pc = p < Np ? p : (Np - 1);
    float x = xyz[3 * pc + 0];
    float y = xyz[3 * pc + 1];
    float z = xyz[3 * pc + 2];
    float r2 = x * x + y * y + z * z;
    float ri = rsqrtf(r2);
    ri = ri * (1.5f - 0.5f * r2 * ri * ri);   // one Newton step
    x *= ri; y *= ri; z *= ri;

    float cm[8], sm[8], zp[8];
    cm[0] = 1.0f; sm[0] = 0.0f;
    #pragma unroll
    for (int m = 1; m < 8; ++m) {
        cm[m] = x * cm[m - 1] - y * sm[m - 1];
        sm[m] = x * sm[m - 1] + y * cm[m - 1];
    }
    zp[0] = 1.0f;
    #pragma unroll
    for (int k = 1; k < 8; ++k) zp[k] = zp[k - 1] * z;

    // feature value for flat index f (compile-time (m,k) after unroll)
    #define FVAL(f) ( (((f) < 36) ? cm[FM(f)] : sm[FM(f)]) * zp[FK(f)] )

    // --- A fragments: 16x4 f32; VGPR0 = {K, K+2}, VGPR1 = {K+1, K+3} ------
    v2f a[16];
    #pragma unroll
    for (int s4 = 0; s4 < 16; ++s4) {
        float lo0 = FVAL(4 * s4 + 0);
        float lo1 = FVAL(4 * s4 + 1);
        float hi0 = FVAL(4 * s4 + 2);
        float hi1 = FVAL(4 * s4 + 3);
        a[s4].x = hb ? hi0 : lo0;
        a[s4].y = hb ? hi1 : lo1;
    }
    #undef FVAL

    // --- 4 output tiles of 16 columns each -------------------------------
    #pragma unroll
    for (int t = 0; t < 4; ++t) {
        const int n = t * 16 + col;                // flat output index of lane
        v8f acc = {0.f, 0.f, 0.f, 0.f, 0.f, 0.f, 0.f, 0.f};
        #pragma unroll
        for (int s4 = 0; s4 < 16; ++s4) {
            v2f b;                                 // B 4x16: rows K / K+2 split
            b.x = sW[(4 * s4 + 2 * h    ) * NOUT + n];
            b.y = sW[(4 * s4 + 2 * h + 1) * NOUT + n];
            acc = __builtin_amdgcn_wmma_f32_16x16x4_f32(
                false, a[s4], false, b, (short)0, acc, false, false);
        }
        // D layout: VGPR r -> point M = r + 8*h, column N = col
        const int4 mt   = sMeta[n];                // {l*l, 2l+1, n-l*l, _}
        const long base = (long)Np * (long)mt.x + (long)mt.z;
        const int  wrow = mt.y;
        const int  pb   = wb + 8 * h;
        #pragma unroll
        for (int r = 0; r < 8; ++r) {
            int pp = pb + r;
            if (pp < Np)
                __builtin_nontemporal_store(acc[r], &out[base + (long)pp * wrow]);
        }
    }
}

// ---------------------------------------------------------------------------
extern "C" void kernel_launch(void* const* d_in, const int* in_sizes, int n_in,
                              void* d_out, int out_size, void* d_ws, size_t ws_size,
                              hipStream_t stream) {
    const float* xyz = (const float*)d_in[0];
    // d_in[1] is l_max; this kernel is specialized for l_max == 7.
    const int Np = in_sizes[0] / 3;

    float* W   = (float*)d_ws;     // 64*64 floats + 64 int4 meta = 17 KB scratch
    float* out = (float*)d_out;

    sh_init_W<<<1, 64, 0, stream>>>(W);

    const int blocks = (Np + 127) / 128;   // 128 points per 256-thread block
    sh_wmma_kernel<<<blocks, 256, 0, stream>>>(xyz, W, out, Np);
}